// HausdorffDistance_81784767250634
// MI455X (gfx1250) — compile-verified
//
#include <hip/hip_runtime.h>
#include <cstdint>

#define INF_F 1e10f
#define NB    16
#define L     64
#define SLICE (L * L)          // 4096
#define VOL   (L * L * L)      // 262144
#define NVOX  (NB * VOL)       // 4194304

typedef __attribute__((ext_vector_type(2))) float v2f;
typedef __attribute__((ext_vector_type(8))) float v8f;

// ---------------------------------------------------------------------------
// CDNA5 async global->LDS copy (ASYNCcnt path, CDNA5 ISA).
// GV mode: 64-bit per-lane global address in a VGPR pair, LDS byte offset in a
// VGPR (low 32 bits of the generic LDS pointer == workgroup-relative offset).
// ---------------------------------------------------------------------------
__device__ __forceinline__ void async_copy_f32(unsigned lds_off, const float* gsrc) {
    asm volatile("global_load_async_to_lds_b32 %0, %1, off"
                 :: "v"(lds_off), "v"(gsrc)
                 : "memory");
}
__device__ __forceinline__ void wait_async() {
    asm volatile("s_wait_asynccnt 0" ::: "memory");
}

// ---------------------------------------------------------------------------
// Kernel 1: binarize + 6-connected boundary (border_value=0 semantics) and
// write EDT seeds (0 on boundary, INF elsewhere) for both masks.
// ---------------------------------------------------------------------------
__global__ void k_boundary(const float* __restrict__ logits,
                           const int*   __restrict__ tgt,
                           unsigned char* __restrict__ pb,
                           unsigned char* __restrict__ tb,
                           float* __restrict__ dp,
                           float* __restrict__ dt) {
    int v = blockIdx.x * blockDim.x + threadIdx.x;
    if (v >= NVOX) return;
    int x = v & 63;
    int y = (v >> 6) & 63;
    int z = (v >> 12) & 63;

    // sigmoid(t) > 0.5  <=>  t > 0
    unsigned pm = logits[v] > 0.0f;
    unsigned tm = tgt[v] > 0;

    unsigned pe = pm, te = tm;
    // 6 face neighbors; outside the volume counts as False (erosion fails).
    {
        int dz[6] = {-1, 1, 0, 0, 0, 0};
        int dy[6] = { 0, 0,-1, 1, 0, 0};
        int dx[6] = { 0, 0, 0, 0,-1, 1};
        #pragma unroll
        for (int k = 0; k < 6; ++k) {
            int zz = z + dz[k], yy = y + dy[k], xx = x + dx[k];
            unsigned inb = ((unsigned)zz < 64u) & ((unsigned)yy < 64u) & ((unsigned)xx < 64u);
            unsigned pn = 0, tn = 0;
            if (inb) {
                int u = v + dz[k] * SLICE + dy[k] * L + dx[k];
                pn = logits[u] > 0.0f;
                tn = tgt[u] > 0;
            }
            pe &= pn;
            te &= tn;
        }
    }
    unsigned pbv = pm & (pe ^ 1u);   // m & ~erode(m)
    unsigned tbv = tm & (te ^ 1u);
    pb[v] = (unsigned char)pbv;
    tb[v] = (unsigned char)tbv;
    dp[v] = pbv ? 0.0f : INF_F;
    dt[v] = tbv ? 0.0f : INF_F;
}

// ---------------------------------------------------------------------------
// Kernel 2: exact 1D squared-EDT pass:  g[i] = min_j (i-j)^2 + f[j]
// pass 0: along x (stride 1), pass 1: along y (stride 64), pass 2: along z
// (stride 4096). Each block owns 4 full lines -> in-place update is safe.
// Lines are staged into LDS with async global->LDS loads (gfx1250 path);
// per-lane addressing performs the strided gather for the y/z passes.
// ---------------------------------------------------------------------------
__global__ void k_edt_pass(float* __restrict__ dp, float* __restrict__ dt, int pass) {
    float* f = (blockIdx.y == 0) ? dp : dt;
    __shared__ float sline[4][L];

    int tx = threadIdx.x;                  // 0..63: position in line
    int ty = threadIdx.y;                  // 0..3 : line within block
    int lineId = blockIdx.x * 4 + ty;      // 0..65535 (NB*64*64 lines)

    long base; int stride;
    if (pass == 0) {                       // line over x: (b,z,y) fixed
        base = (long)lineId * L;           stride = 1;
    } else if (pass == 1) {                // line over y: lineId = (b*64+z)*64 + x
        int bz = lineId >> 6;
        int x  = lineId & 63;
        base = (long)bz * SLICE + x;       stride = L;
    } else {                               // line over z: lineId = b*4096 + (y*64+x)
        int b  = lineId >> 12;
        int yx = lineId & 4095;
        base = (long)b * VOL + yx;         stride = SLICE;
    }

    const float* src = f + base + (long)tx * stride;
    async_copy_f32((unsigned)(uintptr_t)&sline[ty][tx], src);
    wait_async();
    __syncthreads();

    float best = 3.0e38f;
    #pragma unroll
    for (int j = 0; j < L; ++j) {
        int d = tx - j;
        best = fminf(best, sline[ty][j] + (float)(d * d));
    }
    f[base + (long)tx * stride] = best;
}

// ---------------------------------------------------------------------------
// Kernel 3: per-batch directed Hausdorff maxima + nonempty flags.
// hd_pt = max_{pb} sqrt(edt_tb), hd_tp = max_{tb} sqrt(edt_pb).
// ---------------------------------------------------------------------------
__global__ void k_reduce(const unsigned char* __restrict__ pb,
                         const unsigned char* __restrict__ tb,
                         const float* __restrict__ dp,
                         const float* __restrict__ dt,
                         float* __restrict__ hd) {
    __shared__ float s_pt[256], s_tp[256];
    __shared__ int   s_ap[256], s_at[256];
    int b = blockIdx.x, tid = threadIdx.x;
    const unsigned char* pbb = pb + (size_t)b * VOL;
    const unsigned char* tbb = tb + (size_t)b * VOL;
    const float* dpb = dp + (size_t)b * VOL;
    const float* dtb = dt + (size_t)b * VOL;

    float mpt = 0.0f, mtp = 0.0f;
    int ap = 0, at = 0;
    for (int v = tid; v < VOL; v += 256) {
        int p = pbb[v], t = tbb[v];
        ap |= p; at |= t;
        if (p) mpt = fmaxf(mpt, dtb[v]);
        if (t) mtp = fmaxf(mtp, dpb[v]);
    }
    s_pt[tid] = mpt; s_tp[tid] = mtp; s_ap[tid] = ap; s_at[tid] = at;
    __syncthreads();
    for (int o = 128; o > 0; o >>= 1) {
        if (tid < o) {
            s_pt[tid] = fmaxf(s_pt[tid], s_pt[tid + o]);
            s_tp[tid] = fmaxf(s_tp[tid], s_tp[tid + o]);
            s_ap[tid] |= s_ap[tid + o];
            s_at[tid] |= s_at[tid + o];
        }
        __syncthreads();
    }
    if (tid == 0) {
        float h = fmaxf(sqrtf(s_pt[0]), sqrtf(s_tp[0]));
        hd[b] = (s_ap[0] && s_at[0]) ? h : 0.0f;
    }
}

// ---------------------------------------------------------------------------
// Kernel 4: mean over batch via V_WMMA_F32_16X16X4_F32 (single wave32).
//   A[i][k] = hd[4i+k] for i<4 (else 0)   -- documented 16x4 f32 A layout:
//     lane l<16 : row M=l,   holds K=0 (a.x), K=1 (a.y)
//     lane l>=16: row M=l-16, holds K=2 (a.x), K=3 (a.y)
//   B = all-ones (layout independent), C = 0.
//   => D[i][0] = hd[4i]+hd[4i+1]+hd[4i+2]+hd[4i+3], i=0..3.
//   C/D layout: N=0 lives in lane 0, M=r in VGPR r -> lane 0 sums c[0..3].
// ---------------------------------------------------------------------------
__global__ void k_mean(const float* __restrict__ hd, float* __restrict__ out) {
    int l = threadIdx.x;                   // 0..31, one full wave, EXEC all 1s
    int row = (l < 16) ? l : (l - 16);
    int kbase = (l < 16) ? 0 : 2;
    v2f a;
    a.x = (row < 4) ? hd[4 * row + kbase + 0] : 0.0f;
    a.y = (row < 4) ? hd[4 * row + kbase + 1] : 0.0f;
    v2f b;
    b.x = 1.0f;
    b.y = 1.0f;
    v8f c = {};
    c = __builtin_amdgcn_wmma_f32_16x16x4_f32(
        /*neg_a=*/false, a, /*neg_b=*/false, b,
        /*c_mod=*/(short)0, c, /*reuse_a=*/false, /*reuse_b=*/false);
    if (l == 0) {
        float total = c[0] + c[1] + c[2] + c[3];
        out[0] = total * (1.0f / (float)NB);
    }
}

// ---------------------------------------------------------------------------
extern "C" void kernel_launch(void* const* d_in, const int* in_sizes, int n_in,
                              void* d_out, int out_size, void* d_ws, size_t ws_size,
                              hipStream_t stream) {
    const float* logits = (const float*)d_in[0];   // [16,1,64,64,64] fp32
    const int*   tgt    = (const int*)d_in[1];     // [16,1,64,64,64] i32

    char* ws = (char*)d_ws;
    unsigned char* pb = (unsigned char*)ws;                        //  4 MiB
    unsigned char* tb = pb + (size_t)NVOX;                         //  4 MiB
    float* dp = (float*)(ws + 2 * (size_t)NVOX);                   // 16 MiB (EDT of pred boundary)
    float* dt = dp + (size_t)NVOX;                                 // 16 MiB (EDT of target boundary)
    float* hd = dt + (size_t)NVOX;                                 // 64 B per-batch results

    k_boundary<<<NVOX / 256, 256, 0, stream>>>(logits, tgt, pb, tb, dp, dt);

    for (int pass = 0; pass < 3; ++pass)
        k_edt_pass<<<dim3(NB * L * L / 4, 2), dim3(L, 4), 0, stream>>>(dp, dt, pass);

    k_reduce<<<NB, 256, 0, stream>>>(pb, tb, dp, dt, hd);
    k_mean<<<1, 32, 0, stream>>>(hd, (float*)d_out);
}